// SimpleGNN_19937238188631
// MI455X (gfx1250) — compile-verified
//
#include <hip/hip_runtime.h>
#include <math.h>

// ---------------------------------------------------------------------------
// 2-layer GCN for MI455X (gfx1250, wave32).
//   norm = rsqrt(max(deg,1));  h = relu((D_in^-1/2 A D_out^-1/2 X) W1 + b1)
//   out  = (D_in^-1/2 A D_out^-1/2 h) W2 + b2
// GEMMs use native fp32 WMMA (v_wmma_f32_16x16x4_f32) -> full fp32 accuracy.
// Each wave computes a 16x64 output tile (4 WMMA accumulators) so the A
// fragment load is amortized over 4 matrix ops. All per-node scalings are
// fused into the GEMM A-load / epilogue.
// ---------------------------------------------------------------------------

typedef __attribute__((ext_vector_type(2))) float v2f;
typedef __attribute__((ext_vector_type(8))) float v8f;

#define N_NODES 10000
#define N_EDGES 640000
#define IN_DIM  128
#define HID_DIM 512
#define OUT_DIM 512

// ---------------- utility: zero a float region ------------------------------
__global__ void gnn_zero_kernel(float* __restrict__ p, size_t n) {
    size_t i = (size_t)blockIdx.x * blockDim.x + threadIdx.x;
    size_t stride = (size_t)gridDim.x * blockDim.x;
    for (; i < n; i += stride) p[i] = 0.0f;
}

// ---------------- degree histogram via fp32 atomics --------------------------
__global__ void gnn_degree_kernel(const int* __restrict__ src,
                                  const int* __restrict__ dst,
                                  float* __restrict__ deg_out,
                                  float* __restrict__ deg_in, int E) {
    int i = blockIdx.x * blockDim.x + threadIdx.x;
    if (i < E) {
        atomicAdd(&deg_out[src[i]], 1.0f);
        atomicAdd(&deg_in[dst[i]], 1.0f);
    }
}

// ---------------- deg -> rsqrt(max(deg,1)) in place (both arrays at once) ----
__global__ void gnn_norm_kernel(float* __restrict__ deg, int n) {
    int i = blockIdx.x * blockDim.x + threadIdx.x;
    if (i < n) {
        float d = deg[i];
        deg[i] = rsqrtf(fmaxf(d, 1.0f));
    }
}

// ---------------- x = feat * norm_src[row]  (layer-1 input scaling) ----------
__global__ void gnn_scale_kernel(const float* __restrict__ feat,
                                 const float* __restrict__ norm_src,
                                 float* __restrict__ x, int total) {
    int i = blockIdx.x * blockDim.x + threadIdx.x;
    if (i < total) {
        x[i] = feat[i] * norm_src[i >> 7];   // IN_DIM == 128
    }
}

// ---------------- edge scatter-add: agg[dst] += x[src], wave-per-edge --------
// x rows are L2-resident (<= 20.5 MB); atomics stay in L2 -> high throughput.
template <int D>
__global__ __launch_bounds__(256)
void gnn_scatter_kernel(const float* __restrict__ x,
                        const int* __restrict__ src,
                        const int* __restrict__ dst,
                        float* __restrict__ agg, int E) {
    const int wave  = (blockIdx.x * blockDim.x + threadIdx.x) >> 5;
    const int lane  = threadIdx.x & 31;
    const int nwave = (gridDim.x * blockDim.x) >> 5;
    for (int e = wave; e < E; e += nwave) {
        const int s = src[e];
        const int d = dst[e];
        const float* __restrict__ xr = x   + (size_t)s * D;
        float*       __restrict__ ar = agg + (size_t)d * D;
#pragma unroll
        for (int j = 0; j < D / 32; ++j) {
            atomicAdd(&ar[lane + 32 * j], xr[lane + 32 * j]);
        }
    }
}

// ---------------- fused GEMM on the matrix pipe ------------------------------
// out[m,n] = epilogue( sum_k (A[m,k]*rowscale[m]) * W[k,n] + bias[n] )
// epilogue (RELU_POST): relu(.)*postscale[m]   (pre-scales h for layer 2)
// One wave owns a 16x64 output tile: one shared A fragment feeds 4 WMMAs.
// K stepped by 4 with fp32 WMMA. Fragment layouts per CDNA5 ISA 7.12.2:
//   A 16x4 : lane<16 -> K={k0,k0+1}, lane>=16 -> K={k0+2,k0+3}; M = lane&15
//   B 4x16 : mirrored (N = lane&15)
//   C/D    : 8 VGPRs, M = tm + v + 8*(lane>=16), N = tn + (lane&15)
template <int K, int NOUT, bool RELU_POST>
__global__ __launch_bounds__(256)
void gnn_gemm_wmma_kernel(const float* __restrict__ A,
                          const float* __restrict__ rowscale,   // norm_dst
                          const float* __restrict__ W,          // K x NOUT
                          const float* __restrict__ bias,       // NOUT
                          const float* __restrict__ postscale,  // norm_src
                          float* __restrict__ out, int M) {
    const int wave    = (blockIdx.x * blockDim.x + threadIdx.x) >> 5;
    const int lane    = threadIdx.x & 31;
    const int ngroups = NOUT >> 6;                 // 64-wide N groups
    const int tm      = (wave / ngroups) << 4;
    const int tnb     = (wave % ngroups) << 6;
    if (tm >= M) return;                           // wave-uniform: EXEC all-1s

    const int l15  = lane & 15;
    const int half = lane >> 4;                    // K-pair {0,1} vs {2,3}
    const int row  = tm + l15;

    const float rs = rowscale[row];
    const float* __restrict__ arow  = A + (size_t)row * K;
    const float* __restrict__ wbase = W + tnb + l15;

    v8f acc0 = {}, acc1 = {}, acc2 = {}, acc3 = {};
#pragma unroll 4
    for (int k0 = 0; k0 < K; k0 += 4) {
        const int ka = k0 + half * 2;
        float2 a2 = *(const float2*)(arow + ka);           // global_load_b64
        v2f a; a.x = a2.x * rs; a.y = a2.y * rs;           // fold norm_dst
        const float* __restrict__ w0 = wbase + (size_t)ka * NOUT;
        const float* __restrict__ w1 = w0 + NOUT;
        v2f b0; b0.x = w0[0];  b0.y = w1[0];
        v2f b1; b1.x = w0[16]; b1.y = w1[16];
        v2f b2; b2.x = w0[32]; b2.y = w1[32];
        v2f b3; b3.x = w0[48]; b3.y = w1[48];
        acc0 = __builtin_amdgcn_wmma_f32_16x16x4_f32(false, a, false, b0,
                                                     (short)0, acc0, false, false);
        acc1 = __builtin_amdgcn_wmma_f32_16x16x4_f32(false, a, false, b1,
                                                     (short)0, acc1, false, false);
        acc2 = __builtin_amdgcn_wmma_f32_16x16x4_f32(false, a, false, b2,
                                                     (short)0, acc2, false, false);
        acc3 = __builtin_amdgcn_wmma_f32_16x16x4_f32(false, a, false, b3,
                                                     (short)0, acc3, false, false);
    }

    // Epilogue: bias + (optional) relu * postscale, one 16-col strip per acc.
    float ps[8];
#pragma unroll
    for (int v = 0; v < 8; ++v) {
        const int m = tm + v + half * 8;
        ps[v] = RELU_POST ? postscale[m] : 1.0f;
    }
#pragma unroll
    for (int t = 0; t < 4; ++t) {
        const v8f acc = (t == 0) ? acc0 : (t == 1) ? acc1 : (t == 2) ? acc2 : acc3;
        const int ncol = tnb + t * 16 + l15;
        const float bn = bias[ncol];
#pragma unroll
        for (int v = 0; v < 8; ++v) {
            const int m = tm + v + half * 8;
            float val = acc[v] + bn;
            if (RELU_POST) val = fmaxf(val, 0.0f) * ps[v];
            out[(size_t)m * NOUT + ncol] = val;
        }
    }
}

// ---------------------------------------------------------------------------
extern "C" void kernel_launch(void* const* d_in, const int* in_sizes, int n_in,
                              void* d_out, int out_size, void* d_ws, size_t ws_size,
                              hipStream_t stream) {
    const float* feat = (const float*)d_in[0];
    const int*   eidx = (const int*)d_in[1];   // int32 indices (JAX default x64-off)
    const float* W1   = (const float*)d_in[2];
    const float* b1   = (const float*)d_in[3];
    const float* W2   = (const float*)d_in[4];
    const float* b2   = (const float*)d_in[5];
    const int*   src  = eidx;
    const int*   dst  = eidx + N_EDGES;

    // Workspace layout (floats). Zeroed region first (atomics accumulate).
    float* ws       = (float*)d_ws;
    float* deg_out  = ws;                               // [N]   -> becomes norm_src
    float* deg_in   = deg_out + N_NODES;                // [N]   -> becomes norm_dst
    float* agg1     = deg_in  + N_NODES;                // [N,128]
    float* agg2     = agg1 + (size_t)N_NODES * IN_DIM;  // [N,512]
    float* x        = agg2 + (size_t)N_NODES * HID_DIM; // [N,128]
    float* h        = x    + (size_t)N_NODES * IN_DIM;  // [N,512] (pre-scaled by norm_src)
    float* outp     = (float*)d_out;                    // [N,512]

    const size_t zero_n = (size_t)N_NODES * (2 + IN_DIM + HID_DIM);
    gnn_zero_kernel<<<2048, 256, 0, stream>>>(ws, zero_n);

    gnn_degree_kernel<<<(N_EDGES + 255) / 256, 256, 0, stream>>>(
        src, dst, deg_out, deg_in, N_EDGES);

    gnn_norm_kernel<<<(2 * N_NODES + 255) / 256, 256, 0, stream>>>(
        deg_out, 2 * N_NODES);                          // both arrays contiguous

    gnn_scale_kernel<<<(N_NODES * IN_DIM + 255) / 256, 256, 0, stream>>>(
        feat, deg_out, x, N_NODES * IN_DIM);

    // Layer 1: scatter 128-dim, then fused (scale)GEMM(+bias,relu,*norm_src)
    gnn_scatter_kernel<IN_DIM><<<N_EDGES / 8, 256, 0, stream>>>(
        x, src, dst, agg1, N_EDGES);

    {   // waves: (10000/16)=625 M-tiles x (512/64)=8 N-groups = 5000 -> 625 blocks
        const int blocks = (N_NODES / 16) * (HID_DIM / 64) / 8;
        gnn_gemm_wmma_kernel<IN_DIM, HID_DIM, true><<<blocks, 256, 0, stream>>>(
            agg1, deg_in, W1, b1, deg_out, h, N_NODES);
    }

    // Layer 2: scatter 512-dim (h already carries norm_src), fused GEMM+bias
    gnn_scatter_kernel<HID_DIM><<<N_EDGES / 8, 256, 0, stream>>>(
        h, src, dst, agg2, N_EDGES);

    {
        const int blocks = (N_NODES / 16) * (OUT_DIM / 64) / 8;
        gnn_gemm_wmma_kernel<HID_DIM, OUT_DIM, false><<<blocks, 256, 0, stream>>>(
            agg2, deg_in, W2, b2, nullptr, outp, N_NODES);
    }
}